// MultiHeadAttention2Layer_89180700934254
// MI455X (gfx1250) — compile-verified
//
#include <hip/hip_runtime.h>
#include <stdint.h>

#define IN_DIM  128
#define HID     128
#define HEADS   8
#define OUT_DIM 16

typedef float v2f  __attribute__((ext_vector_type(2)));
typedef float v8f  __attribute__((ext_vector_type(8)));
typedef unsigned int u32x4 __attribute__((ext_vector_type(4)));
typedef int   i32x4 __attribute__((ext_vector_type(4)));
typedef int   i32x8 __attribute__((ext_vector_type(8)));

// ---------------------------------------------------------------------------
// TDM: DMA a 128x128 fp32 matrix from global memory into LDS (64 KB).
// D# built per CDNA5 ISA (2D tensor, groups 2/3 unused).
// 6-arg builtin form (amdgpu-toolchain / clang-23).
// ---------------------------------------------------------------------------
__device__ __forceinline__ void tdm_load_w128_to_lds(const float* gptr, unsigned lds_off) {
  uint64_t ga = (uint64_t)(uintptr_t)gptr;
  u32x4 g0;
  g0[0] = 1u;                                   // count=1, user descriptor
  g0[1] = lds_off;                              // lds_addr (bytes)
  g0[2] = (unsigned)(ga & 0xFFFFFFFFu);         // global_addr[31:0]
  g0[3] = (unsigned)((ga >> 32) & 0x01FFFFFFu)  // global_addr[56:32]
        | (2u << 30);                           // type = 2 ("image")
  i32x8 g1;
  g1[0] = (int)(2u << 16);                      // data_size = 2 (4 bytes)
  g1[1] = (int)(128u << 16);                    // tensor_dim0[15:0] = 128
  g1[2] = (int)(128u << 16);                    // dim0 hi = 0 | tensor_dim1[15:0] = 128
  g1[3] = (int)(128u << 16);                    // dim1 hi = 0 | tile_dim0 = 128
  g1[4] = (int)(128u);                          // tile_dim1 = 128 | tile_dim2 = 0
  g1[5] = (int)(128u);                          // tensor_dim0_stride[31:0] = 128 elems
  g1[6] = 0;                                    // stride0 hi | stride1 lo
  g1[7] = 0;                                    // stride1 hi
  i32x4 z4 = {0, 0, 0, 0};
  i32x8 z8 = {0, 0, 0, 0, 0, 0, 0, 0};
  __builtin_amdgcn_tensor_load_to_lds(g0, g1, z4, z4, z8, 0);
  __builtin_amdgcn_s_wait_tensorcnt(0);
}

// ---------------------------------------------------------------------------
// One 16x16 fp32 output tile: A = 16 contiguous rows of 128 (global),
// B = LDS-resident 128x128 weight, cols [col0, col0+16). 32 x WMMA f32 16x16x4.
// ---------------------------------------------------------------------------
__device__ __forceinline__ v8f wmma_tile_16x16_k128(const float* __restrict__ Arows,
                                                    const float* __restrict__ ldsW,
                                                    int col0, int lane) {
  const int m  = lane & 15;           // A row / B col / D col index for this lane
  const int kh = (lane >> 4) << 1;    // lanes 0-15: K offset 0,1 ; lanes 16-31: 2,3
  const float* __restrict__ arow = Arows + m * IN_DIM + kh;
  v8f acc = {};
#pragma unroll
  for (int k0 = 0; k0 < IN_DIM; k0 += 4) {
    v2f a, b;
    a.x = arow[k0 + 0];
    a.y = arow[k0 + 1];
    b.x = ldsW[(k0 + kh + 0) * HID + col0 + m];
    b.y = ldsW[(k0 + kh + 1) * HID + col0 + m];
    acc = __builtin_amdgcn_wmma_f32_16x16x4_f32(false, a, false, b,
                                                (short)0, acc, false, false);
  }
  return acc;
}

// ---------------------------------------------------------------------------
// Y[rows,128] = X[rows,128] @ W[128,128] + bias ; persistent 8-wave blocks.
// ---------------------------------------------------------------------------
__global__ void __launch_bounds__(256)
proj_kernel(const float* __restrict__ X, const float* __restrict__ W,
            const float* __restrict__ bias, float* __restrict__ Y, int nRowTiles) {
  extern __shared__ float ldsW[];               // 128*128 floats = 64 KB
  const int lane = threadIdx.x & 31;
  const int wave = threadIdx.x >> 5;            // 0..7 -> 16-col slice
  if (threadIdx.x < 32)
    tdm_load_w128_to_lds(W, (unsigned)(uintptr_t)&ldsW[0]);
  __syncthreads();

  const int col0  = wave * 16;
  const int n     = lane & 15;
  const int mbase = (lane >> 4) << 3;           // D rows r or r+8 per lane half
  const float bv  = bias[col0 + n];

  for (int rt = blockIdx.x; rt < nRowTiles; rt += gridDim.x) {
    const int row0 = rt << 4;
    v8f acc = wmma_tile_16x16_k128(X + (size_t)row0 * IN_DIM, ldsW, col0, lane);
#pragma unroll
    for (int r = 0; r < 8; ++r)
      Y[(size_t)(row0 + mbase + r) * HID + col0 + n] = acc[r] + bv;
  }
}

// ---------------------------------------------------------------------------
// Fused: E = edge_attr @ We + be (kept in registers) ;
// score[e,h] = sum_d E[e,h*16+d] * K[src,h,d]*Q[dst,h,d]/sqrt(16)
// ---------------------------------------------------------------------------
__global__ void __launch_bounds__(256)
edge_score_kernel(const float* __restrict__ EA, const float* __restrict__ We,
                  const float* __restrict__ be,
                  const float* __restrict__ Qp, const float* __restrict__ Kp,
                  const int* __restrict__ srcI, const int* __restrict__ dstI,
                  float* __restrict__ score, int nEdgeTiles) {
  extern __shared__ float ldsW[];
  const int lane = threadIdx.x & 31;
  const int head = threadIdx.x >> 5;            // wave == head
  if (threadIdx.x < 32)
    tdm_load_w128_to_lds(We, (unsigned)(uintptr_t)&ldsW[0]);
  __syncthreads();

  const int col0  = head * 16;
  const int n     = lane & 15;
  const int mbase = (lane >> 4) << 3;
  const float bv  = be[col0 + n];

  for (int et = blockIdx.x; et < nEdgeTiles; et += gridDim.x) {
    const int e0 = et << 4;
    __builtin_prefetch(EA + (size_t)(e0 + 16) * IN_DIM, 0, 1);
    v8f acc = wmma_tile_16x16_k128(EA + (size_t)e0 * IN_DIM, ldsW, col0, lane);
#pragma unroll
    for (int r = 0; r < 8; ++r) {
      const int e = e0 + mbase + r;
      const int s = srcI[e];
      const int d = dstI[e];
      float g = Kp[(size_t)s * HID + col0 + n] *
                Qp[(size_t)d * HID + col0 + n] * 0.25f;   // 1/sqrt(OUT_DIM)
      float p = (acc[r] + bv) * g;
      p += __shfl_xor(p, 1);
      p += __shfl_xor(p, 2);
      p += __shfl_xor(p, 4);
      p += __shfl_xor(p, 8);                    // reduce 16 lanes of one edge
      if (n == 0)
        score[(size_t)e * HEADS + head] = p;
    }
  }
}

// ---------------------------------------------------------------------------
// Segmented softmax + scatter helpers
// ---------------------------------------------------------------------------
__device__ __forceinline__ unsigned fmap(float f) {     // monotone float->uint
  unsigned u = __float_as_uint(f);
  return (u & 0x80000000u) ? ~u : (u | 0x80000000u);
}
__device__ __forceinline__ float funmap(unsigned u) {
  return (u & 0x80000000u) ? __uint_as_float(u & 0x7FFFFFFFu)
                           : __uint_as_float(~u);
}

__global__ void init_kernel(unsigned* __restrict__ smax, float* __restrict__ denom,
                            float* __restrict__ out, long nNH, long nOut) {
  long i = (long)blockIdx.x * blockDim.x + threadIdx.x;
  long st = (long)gridDim.x * blockDim.x;
  for (long j = i; j < nNH; j += st) { smax[j] = 0x007FFFFFu; denom[j] = 0.0f; }
  for (long j = i; j < nOut; j += st) out[j] = 0.0f;
}

__global__ void segmax_kernel(const float* __restrict__ score, const int* __restrict__ dstI,
                              unsigned* __restrict__ smax, long nEH) {
  long i = (long)blockIdx.x * blockDim.x + threadIdx.x;
  long st = (long)gridDim.x * blockDim.x;
  for (long j = i; j < nEH; j += st) {
    long e = j >> 3; int h = (int)(j & 7);
    atomicMax(&smax[(size_t)dstI[e] * HEADS + h], fmap(score[j]));
  }
}

__global__ void expsum_kernel(float* __restrict__ score, const int* __restrict__ dstI,
                              const unsigned* __restrict__ smax, float* __restrict__ denom,
                              long nEH) {
  long i = (long)blockIdx.x * blockDim.x + threadIdx.x;
  long st = (long)gridDim.x * blockDim.x;
  for (long j = i; j < nEH; j += st) {
    long e = j >> 3; int h = (int)(j & 7);
    size_t k = (size_t)dstI[e] * HEADS + h;
    float ex = __expf(score[j] - funmap(smax[k]));
    score[j] = ex;                                // in-place: score -> ex
    atomicAdd(&denom[k], ex);
  }
}

__global__ void scatter_kernel(const float* __restrict__ ex, const float* __restrict__ denom,
                               const float* __restrict__ Vp,
                               const int* __restrict__ srcI, const int* __restrict__ dstI,
                               float* __restrict__ out, long nEC) {
  long i = (long)blockIdx.x * blockDim.x + threadIdx.x;
  long st = (long)gridDim.x * blockDim.x;
  for (long j = i; j < nEC; j += st) {
    long e = j >> 7; int c = (int)(j & 127); int h = c >> 4;
    int d = dstI[e], s = srcI[e];
    float alpha = ex[(size_t)e * HEADS + h] /
                  (denom[(size_t)d * HEADS + h] + 1e-16f);
    atomicAdd(&out[(size_t)d * HID + c], Vp[(size_t)s * HID + c] * alpha);
  }
}

// ---------------------------------------------------------------------------
extern "C" void kernel_launch(void* const* d_in, const int* in_sizes, int n_in,
                              void* d_out, int out_size, void* d_ws, size_t ws_size,
                              hipStream_t stream) {
  const float* x  = (const float*)d_in[0];
  const float* ea = (const float*)d_in[1];
  const int*   ei = (const int*)  d_in[2];
  const float* Wq = (const float*)d_in[3];
  const float* bq = (const float*)d_in[4];
  const float* Wk = (const float*)d_in[5];
  const float* bk = (const float*)d_in[6];
  const float* We = (const float*)d_in[7];
  const float* be = (const float*)d_in[8];
  const float* Wv = (const float*)d_in[9];
  const float* bv = (const float*)d_in[10];
  float* out = (float*)d_out;

  const long nNodes = in_sizes[0] / IN_DIM;
  const long nEdges = (long)in_sizes[2] / 2;
  const int* srcI = ei;                 // edge_index[0]
  const int* dstI = ei + nEdges;        // edge_index[1]

  char* ws = (char*)d_ws;
  float*    Qp    = (float*)ws;    ws += (size_t)nNodes * HID * sizeof(float);
  float*    Kp    = (float*)ws;    ws += (size_t)nNodes * HID * sizeof(float);
  float*    Vp    = (float*)ws;    ws += (size_t)nNodes * HID * sizeof(float);
  float*    score = (float*)ws;    ws += (size_t)nEdges * HEADS * sizeof(float);
  unsigned* smax  = (unsigned*)ws; ws += (size_t)nNodes * HEADS * sizeof(unsigned);
  float*    denom = (float*)ws;

  const size_t shmem = (size_t)HID * IN_DIM * sizeof(float);   // 64 KB
  const int nRowTiles  = (int)(nNodes / 16);
  const int nEdgeTiles = (int)(nEdges / 16);

  int gProj = nRowTiles < 1024 ? nRowTiles : 1024;
  proj_kernel<<<gProj, 256, shmem, stream>>>(x, Wq, bq, Qp, nRowTiles);
  proj_kernel<<<gProj, 256, shmem, stream>>>(x, Wk, bk, Kp, nRowTiles);
  proj_kernel<<<gProj, 256, shmem, stream>>>(x, Wv, bv, Vp, nRowTiles);

  const long nNH = nNodes * HEADS, nOut = nNodes * (long)HID;
  init_kernel<<<2048, 256, 0, stream>>>(smax, denom, out, nNH, nOut);

  int gEdge = nEdgeTiles < 2048 ? nEdgeTiles : 2048;
  edge_score_kernel<<<gEdge, 256, shmem, stream>>>(ea, We, be, Qp, Kp,
                                                   srcI, dstI, score, nEdgeTiles);

  const long nEH = nEdges * HEADS;
  long gb = (nEH + 255) / 256; if (gb > (1L << 18)) gb = 1L << 18;
  segmax_kernel<<<(unsigned)gb, 256, 0, stream>>>(score, dstI, smax, nEH);
  expsum_kernel<<<(unsigned)gb, 256, 0, stream>>>(score, dstI, smax, denom, nEH);

  const long nEC = nEdges * (long)HID;
  long gc = (nEC + 255) / 256; if (gc > (1L << 20)) gc = 1L << 20;
  scatter_kernel<<<(unsigned)gc, 256, 0, stream>>>(score, denom, Vp,
                                                   srcI, dstI, out, nEC);
}